// GPOnnxWrapper_70574902608252
// MI455X (gfx1250) — compile-verified
//
#include <hip/hip_runtime.h>
#include <hip/hip_bf16.h>
#include <math.h>

#define N_TRAIN 4096
#define N_TEST  32768
#define DIMS    32

typedef __attribute__((ext_vector_type(2))) float v2f;
typedef __attribute__((ext_vector_type(4))) float v4f;
typedef __attribute__((ext_vector_type(8))) float v8f;

// ---------------------------------------------------------------------------
// Pass 1: scale rows by 1/lengthscale (ARD) and compute per-row squared norms.
// ---------------------------------------------------------------------------
__global__ __launch_bounds__(256) void matern_preprocess(
    const float* __restrict__ in, const float* __restrict__ ls,
    float* __restrict__ scaled, float* __restrict__ sqnorm, int nrows) {
  int row = blockIdx.x * blockDim.x + threadIdx.x;
  if (row >= nrows) return;
  const float* r = in + (size_t)row * DIMS;
  float* o = scaled + (size_t)row * DIMS;
  float s = 0.0f;
#pragma unroll
  for (int k = 0; k < DIMS; ++k) {
    float v = r[k] / ls[k];   // ls is lane-uniform -> scalar loads
    o[k] = v;
    s = fmaf(v, v, s);
  }
  sqnorm[row] = s;
}

// ---------------------------------------------------------------------------
// Pass 2: fused Matern-5/2 GEMV.
// One wave32 owns 16 test points. Loop over 256 train tiles of 16 rows:
//   cross(16x16) via 8x V_WMMA_F32_16X16X4_F32 (K=32)
//   d2 -> matern (raw v_sqrt_f32 + v_exp_f32) -> acc += alpha[n]*k
// C/D layout: lane l, VGPR r:  N = l&15,  M = r + 8*(l>>4).
// ---------------------------------------------------------------------------
__global__ __launch_bounds__(256) void matern_gemv(
    const float* __restrict__ Xs, const float* __restrict__ X2,
    const float* __restrict__ Zs, const float* __restrict__ Z2,
    const float* __restrict__ alpha, const float* __restrict__ oscale,
    float* __restrict__ out) {
  const int lane  = threadIdx.x & 31;
  const int wave  = threadIdx.x >> 5;
  const int laneM = lane & 15;
  const int hi    = lane >> 4;                    // 0: rows 0-7, 1: rows 8-15
  const int m_base = (blockIdx.x * 8 + wave) * 16;

  // B fragments for this wave's 16 test columns: invariant over the n loop.
  const float* zrow = Zs + (size_t)(m_base + laneM) * DIMS;
  v2f b[8];
#pragma unroll
  for (int kk = 0; kk < 8; ++kk)
    b[kk] = *(const v2f*)(zrow + 4 * kk + 2 * hi);
  const float z2 = Z2[m_base + laneM];

  const float s5  = 2.23606797749978969f;         // sqrt(5)
  const float c53 = 5.0f / 3.0f;
  // exp(-sqrt5*r) == exp2(k2*r), one v_mul + one v_exp_f32
  const float k2  = -s5 * 1.44269504088896340736f;
  float acc = 0.0f;

  for (int n0 = 0; n0 < N_TRAIN; n0 += 16) {
    const float* xrow = Xs + (size_t)(n0 + laneM) * DIMS;
    // Prefetch the next train tile's row for this lane (global_prefetch_b8).
    if (n0 + 16 < N_TRAIN) __builtin_prefetch(xrow + 16 * DIMS, 0, 3);

    v2f a[8];
#pragma unroll
    for (int kk = 0; kk < 8; ++kk)
      a[kk] = *(const v2f*)(xrow + 4 * kk + 2 * hi);

    v8f c = {};
#pragma unroll
    for (int kk = 0; kk < 8; ++kk)
      c = __builtin_amdgcn_wmma_f32_16x16x4_f32(
          /*neg_a=*/false, a[kk], /*neg_b=*/false, b[kk],
          /*c_mod=*/(short)0, c, /*reuse_a=*/false, /*reuse_b=*/false);

    // Per-lane row metadata: M = 8*hi + r, r = 0..7 (32B-aligned v4 loads).
    const v4f x2a = *(const v4f*)(X2 + n0 + 8 * hi);
    const v4f x2b = *(const v4f*)(X2 + n0 + 8 * hi + 4);
    const v4f ala = *(const v4f*)(alpha + n0 + 8 * hi);
    const v4f alb = *(const v4f*)(alpha + n0 + 8 * hi + 4);

#pragma unroll
    for (int r = 0; r < 8; ++r) {
      const float x2 = (r < 4) ? x2a[r] : x2b[r - 4];
      const float av = (r < 4) ? ala[r] : alb[r - 4];
      float d2 = fmaf(-2.0f, c[r], x2 + z2);
      d2 = fmaxf(d2, 0.0f) + 1e-9f;
      // argument is in [1e-9, O(100)] -> raw v_sqrt_f32 / v_exp_f32 are safe
      const float rd = __builtin_amdgcn_sqrtf(d2);
      const float ev = __builtin_amdgcn_exp2f(k2 * rd);
      const float poly = fmaf(c53 * rd, rd, fmaf(s5, rd, 1.0f));
      acc = fmaf(av, poly * ev, acc);
    }
  }

  // Fold the two M-halves; lanes 0-15 then hold full sums for their column.
  acc += __shfl_xor(acc, 16, 32);
  if (lane < 16) out[m_base + lane] = oscale[0] * acc;
}

// ---------------------------------------------------------------------------
extern "C" void kernel_launch(void* const* d_in, const int* in_sizes, int n_in,
                              void* d_out, int out_size, void* d_ws, size_t ws_size,
                              hipStream_t stream) {
  const float* x       = (const float*)d_in[0];   // (32768, 32)
  const float* train_x = (const float*)d_in[1];   // (4096, 32)
  const float* ls      = (const float*)d_in[2];   // (32,)
  const float* oscale  = (const float*)d_in[3];   // scalar
  const float* alpha   = (const float*)d_in[4];   // (4096,)
  float* out = (float*)d_out;                     // (32768,)

  // Workspace layout (floats): Xs | X2 | Zs | Z2  (~4.9 MB total)
  float* Xs = (float*)d_ws;
  float* X2 = Xs + (size_t)N_TRAIN * DIMS;
  float* Zs = X2 + N_TRAIN;
  float* Z2 = Zs + (size_t)N_TEST * DIMS;

  matern_preprocess<<<(N_TRAIN + 255) / 256, 256, 0, stream>>>(
      train_x, ls, Xs, X2, N_TRAIN);
  matern_preprocess<<<(N_TEST + 255) / 256, 256, 0, stream>>>(
      x, ls, Zs, Z2, N_TEST);

  // 32768 test points / 16 per wave = 2048 waves; 8 waves (256 thr) per block.
  matern_gemv<<<N_TEST / (16 * 8), 256, 0, stream>>>(
      Xs, X2, Zs, Z2, alpha, oscale, out);
}